// MixGCN_59854664237689
// MI455X (gfx1250) — compile-verified
//
#include <hip/hip_runtime.h>
#include <stdint.h>

#define N_USERS   27094
#define M_SPOTS   42852
#define HIDDEN    64
#define NUM_LAYERS 3
#define SLOTS     (NUM_LAYERS + 1)      // 4
#define ROWSTRIDE (SLOTS * HIDDEN)      // 256 floats per row in d_out
#define TPB       256
#define EPB       256                   // edges staged per block

// ---------------------------------------------------------------------------
// CDNA5 async global->LDS helpers (gfx1250). VDST VGPR carries the LDS byte
// offset (low 32 bits of the generic pointer == AS(3) offset); global address
// is a 64-bit VGPR pair. Tracked by ASYNCcnt.
// ---------------------------------------------------------------------------
__device__ __forceinline__ uint32_t lds_off(const void* p) {
    return (uint32_t)(uintptr_t)p;
}
__device__ __forceinline__ void async_load_b128(uint32_t lds, const void* gaddr) {
    asm volatile("global_load_async_to_lds_b128 %0, %1, off"
                 :: "v"(lds), "v"((uint64_t)(uintptr_t)gaddr)
                 : "memory");
}
__device__ __forceinline__ void wait_async0() {
    asm volatile("s_wait_asynccnt 0" ::: "memory");
}

// ---------------------------------------------------------------------------
// Generic fills / copies
// ---------------------------------------------------------------------------
__global__ void k_zero_f(float* __restrict__ p, int n) {
    int i = blockIdx.x * blockDim.x + threadIdx.x;
    if (i < n) p[i] = 0.0f;
}
__global__ void k_zero_i(int* __restrict__ p, int n) {
    int i = blockIdx.x * blockDim.x + threadIdx.x;
    if (i < n) p[i] = 0;
}
__global__ void k_copy_i(const int* __restrict__ a, int* __restrict__ b, int n) {
    int i = blockIdx.x * blockDim.x + threadIdx.x;
    if (i < n) b[i] = a[i];
}

// ---------------------------------------------------------------------------
// Degrees
// ---------------------------------------------------------------------------
__global__ void k_deg_i(const int* __restrict__ eu, const int* __restrict__ es,
                        int* __restrict__ du, int* __restrict__ ds, int E) {
    int e = blockIdx.x * blockDim.x + threadIdx.x;
    if (e < E) {
        atomicAdd(&du[eu[e]], 1);
        atomicAdd(&ds[es[e]], 1);
    }
}
__global__ void k_deg_f(const int* __restrict__ eu, const int* __restrict__ es,
                        float* __restrict__ ud, float* __restrict__ sd, int E) {
    int e = blockIdx.x * blockDim.x + threadIdx.x;
    if (e < E) {
        atomicAdd(&ud[eu[e]], 1.0f);
        atomicAdd(&sd[es[e]], 1.0f);
    }
}

// ---------------------------------------------------------------------------
// Single-block exclusive scan with running carry (n up to ~43K, tiny cost)
// ---------------------------------------------------------------------------
__global__ __launch_bounds__(TPB)
void k_scan(const int* __restrict__ deg, int* __restrict__ off, int n) {
    __shared__ int tile[TPB];
    __shared__ int carry_s;
    const int t = threadIdx.x;
    if (t == 0) carry_s = 0;
    __syncthreads();
    for (int base = 0; base < n; base += TPB) {
        const int i = base + t;
        const int v = (i < n) ? deg[i] : 0;
        tile[t] = v;
        __syncthreads();
        for (int ofs = 1; ofs < TPB; ofs <<= 1) {     // Hillis-Steele inclusive
            int add = (t >= ofs) ? tile[t - ofs] : 0;
            __syncthreads();
            tile[t] += add;
            __syncthreads();
        }
        const int carry = carry_s;
        if (i < n) off[i] = carry + tile[t] - v;      // exclusive
        __syncthreads();
        if (t == TPB - 1) carry_s = carry + tile[TPB - 1];
        __syncthreads();
    }
    if (t == 0) off[n] = carry_s;                     // total sentinel
}

// ---------------------------------------------------------------------------
// CSR fill: edge streams staged via CDNA5 async global->LDS, then one thread
// per edge computes w = 1/sqrt(deg_u*deg_s) and cursor-scatters both CSRs.
// ---------------------------------------------------------------------------
__global__ __launch_bounds__(TPB)
void k_fill(const int* __restrict__ eu, const int* __restrict__ es,
            const int* __restrict__ du, const int* __restrict__ ds,
            int* __restrict__ cur_u, int* __restrict__ cur_s,
            int* __restrict__ csr_u_spot, float* __restrict__ csr_u_w,
            int* __restrict__ csr_s_user, float* __restrict__ csr_s_w,
            int E) {
    __shared__ __attribute__((aligned(16))) int s_u[EPB];
    __shared__ __attribute__((aligned(16))) int s_s[EPB];
    const int t    = threadIdx.x;
    const int base = blockIdx.x * EPB;
    {
        const int grp = t >> 6;           // 0: s_u, 1: s_s, 2..3: idle
        const int k4  = (t & 63) * 4;
        int ge   = base + k4;
        int gmax = (E - 4) & ~3;          // highest aligned in-bounds start
        int gc   = ge > gmax ? gmax : ge;
        if (grp == 0)      async_load_b128(lds_off(&s_u[k4]), eu + gc);
        else if (grp == 1) async_load_b128(lds_off(&s_s[k4]), es + gc);
    }
    wait_async0();
    __syncthreads();

    const int e = base + t;
    if (e < E) {
        const int   u = s_u[t];
        const int   s = s_s[t];
        const float w = 1.0f / sqrtf((float)du[u] * (float)ds[s]);
        const int pu = atomicAdd(&cur_u[u], 1);
        csr_u_spot[pu] = s;  csr_u_w[pu] = w;
        const int ps = atomicAdd(&cur_s[s], 1);
        csr_s_user[ps] = u;  csr_s_w[ps] = w;
    }
}

// ---------------------------------------------------------------------------
// Initialize d_out: slot 0 = input embeddings, slots 1..3 = 0
// ---------------------------------------------------------------------------
__global__ void k_init(const float* __restrict__ uemb, const float* __restrict__ semb,
                       float* __restrict__ spot_out, float* __restrict__ user_out) {
    int i = blockIdx.x * blockDim.x + threadIdx.x;
    const int total_s = M_SPOTS * HIDDEN;
    const int total   = total_s + N_USERS * HIDDEN;
    if (i >= total) return;
    if (i < total_s) {
        int r = i >> 6, d = i & 63;
        float* row = spot_out + (size_t)r * ROWSTRIDE;
        row[d] = semb[i];
        row[HIDDEN + d] = 0.0f; row[2 * HIDDEN + d] = 0.0f; row[3 * HIDDEN + d] = 0.0f;
    } else {
        int j = i - total_s;
        int r = j >> 6, d = j & 63;
        float* row = user_out + (size_t)r * ROWSTRIDE;
        row[d] = uemb[j];
        row[HIDDEN + d] = 0.0f; row[2 * HIDDEN + d] = 0.0f; row[3 * HIDDEN + d] = 0.0f;
    }
}

// ---------------------------------------------------------------------------
// Pull-based propagation (no atomics): 16 lanes per destination row,
// float4 accumulators in registers, one plain 256B store per row.
// ---------------------------------------------------------------------------
__global__ __launch_bounds__(TPB)
void k_gather(const int* __restrict__ off, const int* __restrict__ idx,
              const float* __restrict__ wgt,
              const float* __restrict__ srcprev,   // opposite side, slot l-1
              float* __restrict__ dstnext,         // this side, slot l
              int R) {
    const int lane = threadIdx.x & 15;
    const int grp  = threadIdx.x >> 4;             // 16 rows per block
    const int r    = blockIdx.x * (TPB / 16) + grp;
    if (r >= R) return;
    const int beg  = off[r];
    const int end  = off[r + 1];
    const int foff = lane * 4;
    float4 acc = make_float4(0.f, 0.f, 0.f, 0.f);
    #pragma unroll 4
    for (int j = beg; j < end; ++j) {
        const int   s = idx[j];                    // broadcast within group
        const float w = wgt[j];
        const float4 v = *reinterpret_cast<const float4*>(
            srcprev + (size_t)s * ROWSTRIDE + foff);
        acc.x += w * v.x; acc.y += w * v.y; acc.z += w * v.z; acc.w += w * v.w;
    }
    *reinterpret_cast<float4*>(dstnext + (size_t)r * ROWSTRIDE + foff) = acc;
}

// ---------------------------------------------------------------------------
// Fallback (small workspace): atomic-scatter propagation, async-staged edges
// ---------------------------------------------------------------------------
__global__ void k_inv(const int* __restrict__ eu, const int* __restrict__ es,
                      const float* __restrict__ ud, const float* __restrict__ sd,
                      float* __restrict__ inv, int E) {
    int e = blockIdx.x * blockDim.x + threadIdx.x;
    if (e < E) inv[e] = 1.0f / sqrtf(ud[eu[e]] * sd[es[e]]);
}

__global__ __launch_bounds__(TPB)
void k_prop(const int* __restrict__ eu, const int* __restrict__ es,
            const float* __restrict__ inv,
            const float* __restrict__ uprev, const float* __restrict__ sprev,
            float* __restrict__ unext, float* __restrict__ snext, int E) {
    __shared__ __attribute__((aligned(16))) int   s_u[EPB];
    __shared__ __attribute__((aligned(16))) int   s_s[EPB];
    __shared__ __attribute__((aligned(16))) float s_w[EPB];
    const int t    = threadIdx.x;
    const int base = blockIdx.x * EPB;
    {
        const int grp = t >> 6;
        const int k4  = (t & 63) * 4;
        int ge   = base + k4;
        int gmax = (E - 4) & ~3;
        int gc   = ge > gmax ? gmax : ge;
        if (grp == 0)      async_load_b128(lds_off(&s_u[k4]), eu  + gc);
        else if (grp == 1) async_load_b128(lds_off(&s_s[k4]), es  + gc);
        else if (grp == 2) async_load_b128(lds_off(&s_w[k4]), inv + gc);
    }
    wait_async0();
    __syncthreads();

    const int lane16 = t & 15;
    const int sub    = t >> 4;
    const int foff   = lane16 * 4;
    for (int i = 0; i < EPB / 16; ++i) {
        const int le = i * 16 + sub;
        const int ge = base + le;
        if (ge >= E) continue;
        const int   u = s_u[le];
        const int   s = s_s[le];
        const float w = s_w[le];
        const float4 sv = *reinterpret_cast<const float4*>(
            sprev + (size_t)s * ROWSTRIDE + foff);
        float* up = unext + (size_t)u * ROWSTRIDE + foff;
        atomicAdd(up + 0, sv.x * w); atomicAdd(up + 1, sv.y * w);
        atomicAdd(up + 2, sv.z * w); atomicAdd(up + 3, sv.w * w);
        const float4 uv = *reinterpret_cast<const float4*>(
            uprev + (size_t)u * ROWSTRIDE + foff);
        float* sp = snext + (size_t)s * ROWSTRIDE + foff;
        atomicAdd(sp + 0, uv.x * w); atomicAdd(sp + 1, uv.y * w);
        atomicAdd(sp + 2, uv.z * w); atomicAdd(sp + 3, uv.w * w);
    }
}

// ---------------------------------------------------------------------------
extern "C" void kernel_launch(void* const* d_in, const int* in_sizes, int n_in,
                              void* d_out, int out_size, void* d_ws, size_t ws_size,
                              hipStream_t stream) {
    const float* user_emb  = (const float*)d_in[0];
    const float* spot_emb  = (const float*)d_in[1];
    const int*   edge_user = (const int*)d_in[2];
    const int*   edge_spot = (const int*)d_in[3];
    const int    E = in_sizes[2];

    float* spot_out = (float*)d_out;
    float* user_out = spot_out + (size_t)M_SPOTS * ROWSTRIDE;

    const int eb   = (E + TPB - 1) / TPB;
    const int degN = N_USERS + M_SPOTS;
    const int initN = (N_USERS + M_SPOTS) * HIDDEN;

    // Workspace needed for CSR path (in 4-byte words):
    // deg_u|deg_s (N+M) + off_u (N+1) + off_s (M+1) + cur_u|cur_s (N+M) + 4E
    const size_t csr_words = (size_t)3 * (N_USERS + M_SPOTS) + 2 + (size_t)4 * E;

    if (ws_size >= csr_words * sizeof(int)) {
        // ---------------- CSR pull path (no steady-state atomics) ----------
        int* deg_u = (int*)d_ws;
        int* deg_s = deg_u + N_USERS;
        int* off_u = deg_s + M_SPOTS;              // N+1
        int* off_s = off_u + (N_USERS + 1);        // M+1
        int* cur_u = off_s + (M_SPOTS + 1);
        int* cur_s = cur_u + N_USERS;
        int*   csr_u_spot = cur_s + M_SPOTS;
        float* csr_u_w    = (float*)(csr_u_spot + E);
        int*   csr_s_user = (int*)(csr_u_w + E);
        float* csr_s_w    = (float*)(csr_s_user + E);

        k_zero_i<<<(degN + TPB - 1) / TPB, TPB, 0, stream>>>(deg_u, degN);
        k_deg_i <<<eb, TPB, 0, stream>>>(edge_user, edge_spot, deg_u, deg_s, E);
        k_scan  <<<1, TPB, 0, stream>>>(deg_u, off_u, N_USERS);
        k_scan  <<<1, TPB, 0, stream>>>(deg_s, off_s, M_SPOTS);
        k_copy_i<<<(N_USERS + TPB - 1) / TPB, TPB, 0, stream>>>(off_u, cur_u, N_USERS);
        k_copy_i<<<(M_SPOTS + TPB - 1) / TPB, TPB, 0, stream>>>(off_s, cur_s, M_SPOTS);
        k_fill  <<<(E + EPB - 1) / EPB, TPB, 0, stream>>>(
            edge_user, edge_spot, deg_u, deg_s, cur_u, cur_s,
            csr_u_spot, csr_u_w, csr_s_user, csr_s_w, E);
        k_init  <<<(initN + TPB - 1) / TPB, TPB, 0, stream>>>(
            user_emb, spot_emb, spot_out, user_out);

        for (int l = 1; l <= NUM_LAYERS; ++l) {
            k_gather<<<(N_USERS + 15) / 16, TPB, 0, stream>>>(
                off_u, csr_u_spot, csr_u_w,
                spot_out + (size_t)(l - 1) * HIDDEN,
                user_out + (size_t)l * HIDDEN, N_USERS);
            k_gather<<<(M_SPOTS + 15) / 16, TPB, 0, stream>>>(
                off_s, csr_s_user, csr_s_w,
                user_out + (size_t)(l - 1) * HIDDEN,
                spot_out + (size_t)l * HIDDEN, M_SPOTS);
        }
    } else {
        // ---------------- fallback: atomic-scatter path --------------------
        float* user_degf = (float*)d_ws;
        float* spot_degf = user_degf + N_USERS;
        float* inv_div   = spot_degf + M_SPOTS;

        k_zero_f<<<(degN + TPB - 1) / TPB, TPB, 0, stream>>>(user_degf, degN);
        k_deg_f <<<eb, TPB, 0, stream>>>(edge_user, edge_spot, user_degf, spot_degf, E);
        k_inv   <<<eb, TPB, 0, stream>>>(edge_user, edge_spot, user_degf, spot_degf,
                                         inv_div, E);
        k_init  <<<(initN + TPB - 1) / TPB, TPB, 0, stream>>>(
            user_emb, spot_emb, spot_out, user_out);

        for (int l = 1; l <= NUM_LAYERS; ++l) {
            k_prop<<<(E + EPB - 1) / EPB, TPB, 0, stream>>>(
                edge_user, edge_spot, inv_div,
                user_out + (size_t)(l - 1) * HIDDEN,
                spot_out + (size_t)(l - 1) * HIDDEN,
                user_out + (size_t)l * HIDDEN,
                spot_out + (size_t)l * HIDDEN, E);
        }
    }
}